// MLP1_67482526155035
// MI455X (gfx1250) — compile-verified
//
#include <hip/hip_runtime.h>
#include <math.h>

#define T_LEN 65536
#define ACT_STRIDE 408          // activation row stride (floats)
#define PAN_STRIDE 12           // weight-panel row stride (floats): 16B-aligned rows,
                                // 12*n mod 64 conflict-free over 16 rows
#define PAN_FLOATS (400 * PAN_STRIDE)
#define PAN_BYTES  (PAN_FLOATS * 4)
#define KCHUNK 8                // k columns per async-staged panel
#define NCHUNK (400 / KCHUNK)   // 50 chunks per layer

typedef __attribute__((ext_vector_type(2))) float v2f;
typedef __attribute__((ext_vector_type(8))) float v8f;

// ---------- fast transcendentals (gfx1250 has V_TANH_F32) ----------
#if defined(__has_builtin)
#if __has_builtin(__builtin_amdgcn_tanhf)
#define HAVE_HW_TANH 1
#endif
#if __has_builtin(__builtin_amdgcn_s_wait_asynccnt)
#define HAVE_WAIT_ASYNC 1
#endif
#endif

__device__ inline float fast_tanh(float x) {
#ifdef HAVE_HW_TANH
  return __builtin_amdgcn_tanhf(x);
#else
  return tanhf(x);
#endif
}
__device__ inline float sigm(float x) { return 0.5f * fast_tanh(0.5f * x) + 0.5f; }

// ---------- CDNA5 async global->LDS copy (ASYNCcnt-tracked) ----------
__device__ inline void async_load_b128(unsigned int lds_byte_addr, const float* gsrc) {
  asm volatile("global_load_async_to_lds_b128 %0, %1, off"
               :: "v"(lds_byte_addr), "v"(gsrc)
               : "memory");
}
#ifdef HAVE_WAIT_ASYNC
#define WAIT_ASYNC(n) __builtin_amdgcn_s_wait_asynccnt((unsigned short)(n))
#else
#define WAIT_ASYNC(n) asm volatile("s_wait_asynccnt " #n ::: "memory")
#endif

// =====================================================================
// Kernel 1: 4-layer stacked LSTM scan, one wave32, layers software-
// pipelined across the time loop (layer l at iter t consumes layer l-1's
// output from iter t-1) so the 4 recurrence chains overlap for ILP.
// Lane j computes gate-row j (PyTorch order: rows 0-7=i, 8-15=f,
// 16-23=g, 24-31=o). Gate gather + h broadcast via wave shuffles.
// =====================================================================
__global__ __launch_bounds__(32)
void lstm_scan_kernel(const float* __restrict__ x,       // T x 1 x 8
                      const float* __restrict__ Wih123,  // 3 x 32 x 8
                      const float* __restrict__ Whh123,  // 3 x 32 x 8
                      const float* __restrict__ bih123,  // 3 x 32
                      const float* __restrict__ bhh123,  // 3 x 32
                      const float* __restrict__ Wih4,    // 4 x 8
                      const float* __restrict__ Whh4,    // 4 x 1
                      const float* __restrict__ bih4,    // 4
                      const float* __restrict__ bhh4,    // 4
                      float* __restrict__ seq_out)       // T
{
  const int j = threadIdx.x;      // 0..31
  const int e = j & 7;            // state element this lane owns

  float wih[3][8], whh[3][8], bias[3];
#pragma unroll
  for (int l = 0; l < 3; ++l) {
#pragma unroll
    for (int k = 0; k < 8; ++k) {
      wih[l][k] = Wih123[(l * 32 + j) * 8 + k];
      whh[l][k] = Whh123[(l * 32 + j) * 8 + k];
    }
    bias[l] = bih123[l * 32 + j] + bhh123[l * 32 + j];
  }
  const int j4 = j & 3;
  float wih4[8];
#pragma unroll
  for (int k = 0; k < 8; ++k) wih4[k] = Wih4[j4 * 8 + k];
  const float whh4  = Whh4[j4];
  const float bias4 = bih4[j4] + bhh4[j4];

  float hb[3][8];  // broadcast h per layer (also the next layer's input buffer)
  float c[3];      // cell element e, replicated per 8-lane group
#pragma unroll
  for (int l = 0; l < 3; ++l) {
#pragma unroll
    for (int k = 0; k < 8; ++k) hb[l][k] = 0.f;
    c[l] = 0.f;
  }
  float h3 = 0.f, c3v = 0.f;

  for (int t = 0; t < T_LEN + 3; ++t) {
    const bool d0 = (t < T_LEN);
    const bool d1 = (t >= 1) && (t <= T_LEN);
    const bool d2 = (t >= 2) && (t <= T_LEN + 1);
    const bool d3 = (t >= 3);

    // phase 1: all gate pre-activations from OLD states (4 independent chains)
    float g0 = 0.f, g1 = 0.f, g2 = 0.f, g4 = 0.f;
    if (d0) {
      const float4* xp = (const float4*)(x + (size_t)t * 8);
      float4 xa = xp[0], xb = xp[1];
      float xv[8] = {xa.x, xa.y, xa.z, xa.w, xb.x, xb.y, xb.z, xb.w};
      g0 = bias[0];
#pragma unroll
      for (int k = 0; k < 8; ++k) g0 += wih[0][k] * xv[k];
#pragma unroll
      for (int k = 0; k < 8; ++k) g0 += whh[0][k] * hb[0][k];
    }
    if (d1) {
      g1 = bias[1];
#pragma unroll
      for (int k = 0; k < 8; ++k) g1 += wih[1][k] * hb[0][k];
#pragma unroll
      for (int k = 0; k < 8; ++k) g1 += whh[1][k] * hb[1][k];
    }
    if (d2) {
      g2 = bias[2];
#pragma unroll
      for (int k = 0; k < 8; ++k) g2 += wih[2][k] * hb[1][k];
#pragma unroll
      for (int k = 0; k < 8; ++k) g2 += whh[2][k] * hb[2][k];
    }
    if (d3) {
      g4 = bias4 + whh4 * h3;
#pragma unroll
      for (int k = 0; k < 8; ++k) g4 += wih4[k] * hb[2][k];
    }

    // phase 2: state updates (gather 4 gates per element via shuffles)
    float hn0 = 0.f, hn1 = 0.f, hn2 = 0.f;
    if (d0) {
      float ig = __shfl(g0, e), fg = __shfl(g0, e + 8);
      float gg = __shfl(g0, e + 16), og = __shfl(g0, e + 24);
      float cn = sigm(fg) * c[0] + sigm(ig) * fast_tanh(gg);
      c[0] = cn;
      hn0 = sigm(og) * fast_tanh(cn);
    }
    if (d1) {
      float ig = __shfl(g1, e), fg = __shfl(g1, e + 8);
      float gg = __shfl(g1, e + 16), og = __shfl(g1, e + 24);
      float cn = sigm(fg) * c[1] + sigm(ig) * fast_tanh(gg);
      c[1] = cn;
      hn1 = sigm(og) * fast_tanh(cn);
    }
    if (d2) {
      float ig = __shfl(g2, e), fg = __shfl(g2, e + 8);
      float gg = __shfl(g2, e + 16), og = __shfl(g2, e + 24);
      float cn = sigm(fg) * c[2] + sigm(ig) * fast_tanh(gg);
      c[2] = cn;
      hn2 = sigm(og) * fast_tanh(cn);
    }
    if (d3) {
      float ig = __shfl(g4, 0), fg = __shfl(g4, 1);
      float gg = __shfl(g4, 2), og = __shfl(g4, 3);
      c3v = sigm(fg) * c3v + sigm(ig) * fast_tanh(gg);
      h3  = sigm(og) * fast_tanh(c3v);
      if (j == 0) seq_out[t - 3] = h3;
    }

    // phase 3: rebroadcast new h vectors for next iteration
    if (d0) {
#pragma unroll
      for (int k = 0; k < 8; ++k) hb[0][k] = __shfl(hn0, k);
    }
    if (d1) {
#pragma unroll
      for (int k = 0; k < 8; ++k) hb[1][k] = __shfl(hn1, k);
    }
    if (d2) {
#pragma unroll
      for (int k = 0; k < 8; ++k) hb[2][k] = __shfl(hn2, k);
    }
  }
}

// =====================================================================
// Kernel 2: fused MLP via V_WMMA_F32_16X16X4_F32, one wave32 per 16-row
// tile. All 25 output-tile accumulators (200 VGPRs) live in registers,
// so each layer makes ONE pass over the activation tile and the tile is
// updated in place in a single LDS buffer. Layers 2/3 stream their
// 400x400 weights through a double-buffered LDS k-panel staged with
// global_load_async_to_lds_b128 (ASYNCcnt), keeping global loads out of
// the wmma dependence chain.
//
// f32 WMMA fragment mapping (ISA 7.12.2):
//   A (16x4):  a.x = A[row][k0+koff], a.y = A[row][k0+koff+1],
//              row = lane&15, koff = (lane>=16)?2:0
//   B (4x16):  bb.x = B[k0+koff][col] = W[col][k0+koff], col = lane&15
//   C/D 16x16: reg r holds row (r + (lane>=16?8:0)), col = lane&15
// =====================================================================
__device__ inline v8f wmma4(v2f a, v2f b, v8f c) {
  return __builtin_amdgcn_wmma_f32_16x16x4_f32(
      /*neg_a=*/false, a, /*neg_b=*/false, b,
      /*c_mod=*/(short)0, c, /*reuse_a=*/false, /*reuse_b=*/false);
}

// stage k-chunk c of W (400 rows x KCHUNK cols) into panel p: 800 16B units
__device__ inline void stage_panel(unsigned int wpan_base, int p,
                                   const float* __restrict__ W, int c, int lane) {
  const int k0 = c * KCHUNK;
#pragma unroll
  for (int it = 0; it < 25; ++it) {          // 800 units / 32 lanes
    int i = it * 32 + lane;
    int n = i >> 1, sub = i & 1;
    unsigned int ldsa =
        wpan_base + (unsigned int)(p * PAN_BYTES + n * (PAN_STRIDE * 4) + sub * 16);
    async_load_b128(ldsa, W + (size_t)n * 400 + k0 + sub * 4);
  }
}

__device__ inline void mlp_layer400_async(float* __restrict__ act,   // LDS 16 x 400
                                          float* __restrict__ wpan,  // LDS 2 panels
                                          const float* __restrict__ W,
                                          const float* __restrict__ bias,
                                          int lane, int row, int koff, int rbase) {
  const unsigned int wpan_base = (unsigned int)(uintptr_t)&wpan[0];

  v8f acc[25];
#pragma unroll
  for (int u = 0; u < 25; ++u) {
    float bv = bias[u * 16 + row];
#pragma unroll
    for (int r8 = 0; r8 < 8; ++r8) acc[u][r8] = bv;
  }

  const float* arow = act + row * ACT_STRIDE + koff;

  stage_panel(wpan_base, 0, W, 0, lane);
  for (int cch = 0; cch < NCHUNK; ++cch) {
    const int p = cch & 1;
    if (cch + 1 < NCHUNK) {
      stage_panel(wpan_base, p ^ 1, W, cch + 1, lane);
      WAIT_ASYNC(25);                        // chunk cch landed; next stays in flight
    } else {
      WAIT_ASYNC(0);
    }
    const float* pan = wpan + p * PAN_FLOATS;
#pragma unroll
    for (int ks = 0; ks < KCHUNK; ks += 4) { // 2 k-steps per chunk
      v2f a;
      a.x = arow[cch * KCHUNK + ks];
      a.y = arow[cch * KCHUNK + ks + 1];
      const int kl = ks + koff;
#pragma unroll
      for (int u = 0; u < 25; ++u) {
        const int n = u * 16 + row;
        v2f bb;
        bb.x = pan[n * PAN_STRIDE + kl];
        bb.y = pan[n * PAN_STRIDE + kl + 1];
        acc[u] = wmma4(a, bb, acc[u]);
      }
    }
  }

  // leaky_relu + in-place write-back (input fully consumed above)
#pragma unroll
  for (int u = 0; u < 25; ++u)
#pragma unroll
    for (int r8 = 0; r8 < 8; ++r8) {
      float v = acc[u][r8];
      v = v > 0.f ? v : 0.01f * v;
      act[(rbase + r8) * ACT_STRIDE + u * 16 + row] = v;
    }
}

__global__ __launch_bounds__(32)
void mlp_fused_kernel(const float* __restrict__ seq,      // T (LSTM output)
                      const float* __restrict__ history,  // T x 56
                      const float* __restrict__ W1, const float* __restrict__ b1,
                      const float* __restrict__ W2, const float* __restrict__ b2,
                      const float* __restrict__ W3, const float* __restrict__ b3,
                      const float* __restrict__ Wo, const float* __restrict__ bo,
                      float* __restrict__ out)             // T
{
  __shared__ __align__(16) float act[16 * ACT_STRIDE];
  __shared__ __align__(16) float wpan[2 * PAN_FLOATS];

  const int lane  = threadIdx.x;
  const int m0    = blockIdx.x * 16;
  const int row   = lane & 15;
  const int koff  = (lane >> 4) << 1;   // 0 or 2
  const int rbase = (lane >> 4) << 3;   // 0 or 8

  // ---- stage combined = [seq | history], K padded 57 -> 64 with zeros
  for (int idx = lane; idx < 16 * 64; idx += 32) {
    int r = idx >> 6, k = idx & 63;
    int grow = m0 + r;
    float v = 0.f;
    if (k == 0)       v = seq[grow];
    else if (k < 57)  v = history[(size_t)grow * 56 + (k - 1)];
    act[r * ACT_STRIDE + k] = v;
  }
  __syncthreads();

  // ---- layer 1: (16x57) @ W1^T + b1, leaky_relu, in place (K padded, B guarded)
  {
    v8f acc[25];
#pragma unroll
    for (int u = 0; u < 25; ++u) {
      float bv = b1[u * 16 + row];
#pragma unroll
      for (int r8 = 0; r8 < 8; ++r8) acc[u][r8] = bv;
    }
    const float* arow = act + row * ACT_STRIDE + koff;
    for (int k0 = 0; k0 < 64; k0 += 4) {
      v2f a;
      a.x = arow[k0];
      a.y = arow[k0 + 1];
      int kk = k0 + koff;
#pragma unroll
      for (int u = 0; u < 25; ++u) {
        int n = u * 16 + row;
        v2f bb;
        bb.x = (kk     < 57) ? W1[n * 57 + kk]     : 0.f;
        bb.y = (kk + 1 < 57) ? W1[n * 57 + kk + 1] : 0.f;
        acc[u] = wmma4(a, bb, acc[u]);
      }
    }
#pragma unroll
    for (int u = 0; u < 25; ++u)
#pragma unroll
      for (int r8 = 0; r8 < 8; ++r8) {
        float v = acc[u][r8];
        v = v > 0.f ? v : 0.01f * v;
        act[(rbase + r8) * ACT_STRIDE + u * 16 + row] = v;
      }
  }
  __syncthreads();

  mlp_layer400_async(act, wpan, W2, b2, lane, row, koff, rbase);   // layer 2
  __syncthreads();
  mlp_layer400_async(act, wpan, W3, b3, lane, row, koff, rbase);   // layer 3
  __syncthreads();

  // ---- output: relu(h3 @ Wo^T + bo); 2 lanes per row, 200 MACs each
  {
    const int half = lane >> 4;
    const float* hr = act + row * ACT_STRIDE + half * 200;
    const float* wo = Wo + half * 200;
    float s = 0.f;
    for (int k = 0; k < 200; ++k) s += hr[k] * wo[k];
    s += __shfl_xor(s, 16);
    s += bo[0];
    s = s > 0.f ? s : 0.f;
    if (lane < 16) out[m0 + row] = s;
  }
}

// =====================================================================
extern "C" void kernel_launch(void* const* d_in, const int* in_sizes, int n_in,
                              void* d_out, int out_size, void* d_ws, size_t ws_size,
                              hipStream_t stream) {
  const float* x       = (const float*)d_in[0];
  const float* history = (const float*)d_in[1];
  const float* Wih123  = (const float*)d_in[2];
  const float* Whh123  = (const float*)d_in[3];
  const float* bih123  = (const float*)d_in[4];
  const float* bhh123  = (const float*)d_in[5];
  const float* Wih4    = (const float*)d_in[6];
  const float* Whh4    = (const float*)d_in[7];
  const float* bih4    = (const float*)d_in[8];
  const float* bhh4    = (const float*)d_in[9];
  const float* W1      = (const float*)d_in[10];
  const float* b1      = (const float*)d_in[11];
  const float* W2      = (const float*)d_in[12];
  const float* b2      = (const float*)d_in[13];
  const float* W3      = (const float*)d_in[14];
  const float* b3      = (const float*)d_in[15];
  const float* Wo      = (const float*)d_in[16];
  const float* bo      = (const float*)d_in[17];

  float* seq = (float*)d_ws;    // T floats of LSTM output
  float* out = (float*)d_out;

  lstm_scan_kernel<<<1, 32, 0, stream>>>(x, Wih123, Whh123, bih123, bhh123,
                                         Wih4, Whh4, bih4, bhh4, seq);
  mlp_fused_kernel<<<T_LEN / 16, 32, 0, stream>>>(seq, history, W1, b1, W2, b2,
                                                  W3, b3, Wo, bo, out);
}